// AveragedIWE_11879879544375
// MI455X (gfx1250) — compile-verified
//
#include <hip/hip_runtime.h>
#include <cstdint>

#define IW_H 128
#define IW_W 128
#define IW_R (IW_H * IW_W)                     // 16384
#define BITMAP_WORDS ((2u * IW_R * IW_R) / 32) // 16,777,216 words = 64 MB

// ---------------- init: zero output image & contribution counters ----------------
__global__ __launch_bounds__(256)
void iwe_init_kernel(float* __restrict__ out, float* __restrict__ contrib, int n) {
    int i = blockIdx.x * blockDim.x + threadIdx.x;
    if (i < n) { out[i] = 0.0f; contrib[i] = 0.0f; }
}

// ---------------- clear 64MB dedup bitmap (reused per batch, L2-resident) --------
__global__ __launch_bounds__(256)
void iwe_clear_kernel(uint4* __restrict__ bm, int n4) {
    int stride = gridDim.x * blockDim.x;
    for (int i = blockIdx.x * blockDim.x + threadIdx.x; i < n4; i += stride)
        bm[i] = make_uint4(0u, 0u, 0u, 0u);
}

// ---------------- per-batch event warp + scatter + exact dedup -------------------
// Events streamed via CDNA5 async global->LDS loads, double-buffered per wave.
__global__ __launch_bounds__(256)
void iwe_events_kernel(const float* __restrict__ ev,      // N*4 floats (ts,y,x,p)
                       const float* __restrict__ flow0,   // R floats (x-flow, ch0)
                       const float* __restrict__ flow1,   // R floats (y-flow, ch1)
                       float* __restrict__ out,           // 2*R (pos image, neg image)
                       float* __restrict__ contrib,       // 2*R
                       unsigned* __restrict__ bitmap,     // BITMAP_WORDS
                       int N) {
    __shared__ float4 buf[2][8][32];                      // [phase][wave-in-block][lane]

    const int lane = threadIdx.x & 31;
    const int wid  = threadIdx.x >> 5;
    const int wavesPerBlock = blockDim.x >> 5;
    const int waveGlobal = blockIdx.x * wavesPerBlock + wid;
    const int numWaves   = gridDim.x * wavesPerBlock;
    const int numChunks  = (N + 31) >> 5;                 // 32 events per wave-chunk

    if (waveGlobal >= numChunks) return;

    const unsigned ldsA = (unsigned)(uintptr_t)(&buf[0][wid][lane]);
    const unsigned ldsB = (unsigned)(uintptr_t)(&buf[1][wid][lane]);

    // issue one 16B event record per lane into LDS via async DMA-style load
    auto issue = [&](int chunk, unsigned ldsDst) {
        int e = (chunk << 5) + lane;
        if (e >= N) e = N - 1;                            // clamp tail (avoid EXEC games)
        int voff = e << 4;                                // byte offset into event stream
        asm volatile("global_load_async_to_lds_b128 %0, %1, %2"
                     :: "v"(ldsDst), "v"(voff), "s"(ev)
                     : "memory");
    };

    issue(waveGlobal, ldsA);                              // prime the pipeline
    int cur = 0;
    for (int c = waveGlobal; c < numChunks; c += numWaves) {
        int cn = c + numWaves;
        if (cn < numChunks) {
            issue(cn, cur ? ldsA : ldsB);                 // prefetch next chunk
            asm volatile("s_wait_asynccnt 1" ::: "memory"); // current chunk landed
        } else {
            asm volatile("s_wait_asynccnt 0" ::: "memory");
        }

        float4 e4 = cur ? buf[1][wid][lane] : buf[0][wid][lane];
        int eid = (c << 5) + lane;
        if (eid < N) {
            float ts = e4.x, y = e4.y, x = e4.z, p = e4.w;
            int lin = (int)y * IW_W + (int)x;
            float fx = flow0[lin];                        // flow channel 0 -> x motion
            float fy = flow1[lin];                        // flow channel 1 -> y motion
            float t1 = __fadd_rn(1.0f, -ts);
            // unfused mul,mul,add to match XLA rounding at half-integer boundaries
            float wy = __fadd_rn(y, __fmul_rn(__fmul_rn(t1, fy), 128.0f));
            float wx = __fadd_rn(x, __fmul_rn(__fmul_rn(t1, fx), 128.0f));
            float iyf = rintf(wy);                        // v_rndne: round half to even
            float ixf = rintf(wx);
            if (iyf >= 0.0f && iyf < (float)IW_H && ixf >= 0.0f && ixf < (float)IW_W) {
                int fw = (int)iyf * IW_W + (int)ixf;
                int cb = (p > 0.0f) ? 0 : 1;              // 0 = positive, 1 = negative
                atomicAdd(&out[cb * IW_R + fw], 1.0f);    // weights * pol_mask == 1.0
                // exact dedup of (pol, lin, fw): 29-bit key into bitmap
                unsigned key  = ((unsigned)cb << 28) | ((unsigned)lin << 14) | (unsigned)fw;
                unsigned word = key >> 5;
                unsigned m    = 1u << (key & 31u);
                unsigned old  = atomicOr(&bitmap[word], m);
                if ((old & m) == 0u)
                    atomicAdd(&contrib[cb * IW_R + fw], 1.0f);
            }
        }
        cur ^= 1;
    }
}

// ---------------- divide accumulated image by distinct-source counts -------------
__global__ __launch_bounds__(256)
void iwe_div_kernel(float* __restrict__ out, const float* __restrict__ contrib, int n) {
    int i = blockIdx.x * blockDim.x + threadIdx.x;
    if (i < n) {
        float c = contrib[i];
        if (c > 0.0f) out[i] = out[i] / c;                // max(c,1)==c when c>0
    }
}

extern "C" void kernel_launch(void* const* d_in, const int* in_sizes, int n_in,
                              void* d_out, int out_size, void* d_ws, size_t ws_size,
                              hipStream_t stream) {
    const float* flow   = (const float*)d_in[0];          // B*2*R
    const float* events = (const float*)d_in[1];          // B*N*4
    // d_in[2] (pol_mask) deliberately unused: it equals (p>0, p<=0), derived from p.

    const int B = in_sizes[0] / (2 * IW_R);
    const int N = in_sizes[1] / (B * 4);

    float*    out     = (float*)d_out;                    // B*2*R
    float*    contrib = (float*)d_ws;                     // B*2*R floats = 1 MB
    size_t    cbytes  = ((size_t)B * 2 * IW_R * sizeof(float) + 255) & ~(size_t)255;
    unsigned* bitmap  = (unsigned*)((char*)d_ws + cbytes);// 64 MB, reused per batch

    const int total = B * 2 * IW_R;                       // 262144
    iwe_init_kernel<<<(total + 255) / 256, 256, 0, stream>>>(out, contrib, total);

    const int n4 = (int)(BITMAP_WORDS / 4);
    for (int b = 0; b < B; ++b) {
        iwe_clear_kernel<<<2048, 256, 0, stream>>>((uint4*)bitmap, n4);
        iwe_events_kernel<<<512, 256, 0, stream>>>(
            events + (size_t)b * N * 4,
            flow + (size_t)b * 2 * IW_R,
            flow + (size_t)b * 2 * IW_R + IW_R,
            out + (size_t)b * 2 * IW_R,
            contrib + (size_t)b * 2 * IW_R,
            bitmap, N);
    }

    iwe_div_kernel<<<(total + 255) / 256, 256, 0, stream>>>(out, contrib, total);
}